// TestModel_38465727103476
// MI455X (gfx1250) — compile-verified
//
#include <hip/hip_runtime.h>
#include <hip/hip_bf16.h>

// Sparse 3x3x3 conv as gather-GEMM: out[N,16] = G[N,54] @ Wflat[54,16]
// G[n, 2k+c] = (nbr[n,k] >= 0) ? features[nbr[n,k], c] : 0
// One wave computes a 16x16 output tile via 14x V_WMMA_F32_16X16X4_F32.

typedef __attribute__((ext_vector_type(2))) float v2f;
typedef __attribute__((ext_vector_type(8))) float v8f;

#define NKERN 27              // kernel offsets
#define CIN   2
#define COUT  16
#define KKTOT (NKERN * CIN)   // 54 reduction length
#define STEPS 14              // ceil(54/4) -> padded to 56
#define WAVES_PER_BLOCK 4

__global__ __launch_bounds__(WAVES_PER_BLOCK * 32)
void spconv_wmma_f32_kernel(const float* __restrict__ features,
                            const float* __restrict__ weights,   // [27,2,16] flat
                            const int*   __restrict__ nbr,       // [N,27]
                            float*       __restrict__ out,       // [N,16]
                            int n)
{
    const int lane = threadIdx.x & 31;
    const int wave = threadIdx.x >> 5;
    const int tile = blockIdx.x * WAVES_PER_BLOCK + wave;   // 16-voxel tile
    const int base = tile * 16;
    if (base >= n) return;                                  // wave-uniform exit

    const int g = lane >> 4;     // half-wave: selects K sub-pair / D row block
    const int m = lane & 15;     // A row within tile == B/D column (channel)

    // A-row this lane gathers for (clamped so tail tiles stay in-bounds).
    int row = base + m;
    if (row > n - 1) row = n - 1;

    // ---- Preload neighbor indices: lane needs k = 2*i + g for each step ----
    int idxv[STEPS];
#pragma unroll
    for (int i = 0; i < STEPS; ++i) {
        const int k = 2 * i + g;                 // neighbor offset id
        idxv[i] = (k < NKERN) ? nbr[(long)row * NKERN + k] : -1;
    }

    // ---- Preload B fragments: lane supplies Wflat[4i+2g][m], Wflat[4i+2g+1][m]
    // Wflat[kk][d] = weights[kk*16 + d] with kk = k*CIN + c  (rows >= 54 are pad)
    v2f bfrag[STEPS];
#pragma unroll
    for (int i = 0; i < STEPS; ++i) {
        const int kk0 = 4 * i + 2 * g;
        float bx = (kk0     < KKTOT) ? weights[kk0 * COUT + m]       : 0.0f;
        float by = (kk0 + 1 < KKTOT) ? weights[(kk0 + 1) * COUT + m] : 0.0f;
        bfrag[i] = (v2f){bx, by};
    }

    // ---- Main loop: masked float2 gather -> f32 WMMA (K=4 per step) ----
    v8f c = {};
#pragma unroll
    for (int i = 0; i < STEPS; ++i) {
        const int idx = idxv[i];
        v2f a = (v2f){0.0f, 0.0f};
        if (idx >= 0) {
            a = *(const v2f*)(features + ((long)idx << 1));  // both input channels
        }
        // 8 args: (neg_a, A, neg_b, B, c_mod, C, reuse_a, reuse_b)
        c = __builtin_amdgcn_wmma_f32_16x16x4_f32(
                false, a, false, bfrag[i], (short)0, c, false, false);
    }

    // ---- Store D: lane L, VGPR r holds out[base + r + 8g][m] ----
#pragma unroll
    for (int r = 0; r < 8; ++r) {
        const int orow = base + r + 8 * g;
        if (orow < n) {
            out[(long)orow * COUT + m] = c[r];
        }
    }
}

extern "C" void kernel_launch(void* const* d_in, const int* in_sizes, int n_in,
                              void* d_out, int out_size, void* d_ws, size_t ws_size,
                              hipStream_t stream) {
    const float* features = (const float*)d_in[0];   // [N,2] f32
    const float* weights  = (const float*)d_in[1];   // [27,2,16] f32
    const int*   nbr      = (const int*)d_in[2];     // [N,27] i32
    float* out = (float*)d_out;                      // [N,16] f32

    const int n = in_sizes[0] / CIN;
    const int tiles = (n + 15) / 16;
    const int blocks = (tiles + WAVES_PER_BLOCK - 1) / WAVES_PER_BLOCK;

    spconv_wmma_f32_kernel<<<blocks, dim3(WAVES_PER_BLOCK * 32), 0, stream>>>(
        features, weights, nbr, out, n);
}